// Net_78486232367475
// MI455X (gfx1250) — compile-verified
//
#include <hip/hip_runtime.h>

typedef __attribute__((ext_vector_type(2))) float v2f;
typedef __attribute__((ext_vector_type(8))) float v8f;

#define NB  2
#define C   16
#define T   16
#define H   64
#define W   64
#define K3  27
#define CO  81        // 3*K3 offset channels
#define RK  432       // C*K3 reduction length
#define HW  4096      // H*W
#define THW 65536     // T*H*W
#define P   (NB*THW)  // 131072 flattened output positions

// workspace layout (bytes)
#define OFF_BYTES   ((size_t)P * CO * 4)        // 42,467,328
#define XT_OFFSET   OFF_BYTES
#define XT_BYTES    ((size_t)NB * THW * C * 4)  // 8,388,608
#define WOT_OFFSET  (XT_OFFSET + XT_BYTES)
#define WOT_BYTES   ((size_t)CO * RK * 4)       // 139,968
#define WT_OFFSET   (WOT_OFFSET + WOT_BYTES)

// ---------------------------------------------------------------------------
// Prep: channel-innermost transposes so per-lane (c0, c0+1) pairs are b64.
//   xT[n,s,c]      = x[n,c,s]            (s = t*HW + h*W + w)
//   w_offT[o,r,c]  = w_off[o,c,r]        (r = kt*9+kh*3+kw)
//   wT[o,r,c]      = w[o,c,r]
// ---------------------------------------------------------------------------
__global__ __launch_bounds__(256) void prep(const float* __restrict__ x,
                                            const float* __restrict__ w_off,
                                            const float* __restrict__ wmat,
                                            float* __restrict__ xT,
                                            float* __restrict__ w_offT,
                                            float* __restrict__ wT)
{
    const int idx = blockIdx.x * 256 + threadIdx.x;   // [0, 2^21)

    // x transpose: idx = n<<20 | s<<4 | c
    {
        const int n = idx >> 20;
        const int s = (idx >> 4) & (THW - 1);
        const int c = idx & 15;
        xT[idx] = x[(n << 20) | (c << 16) | s];
    }
    // w_off transpose: 81*432 = 34992 elements
    if (idx < CO * RK) {
        const int o = idx / RK;
        const int rem = idx - o * RK;
        const int r = rem >> 4;
        const int c = rem & 15;
        w_offT[idx] = w_off[o * RK + c * K3 + r];
    }
    // w transpose: 16*432 = 6912 elements
    if (idx < 16 * RK) {
        const int o = idx / RK;
        const int rem = idx - o * RK;
        const int r = rem >> 4;
        const int c = rem & 15;
        wT[idx] = wmat[o * RK + c * K3 + r];
    }
}

// ---------------------------------------------------------------------------
// Kernel 1: offset field GEMM, M=16 positions x N=96 (6 col tiles, one wave).
// K=432 in steps of 4; A gathered once per step (b64) and reused by 6 WMMAs.
// ---------------------------------------------------------------------------
__global__ __launch_bounds__(256) void offset_gemm(const float* __restrict__ xT,
                                                   const float* __restrict__ w_offT,
                                                   const float* __restrict__ b_off,
                                                   float* __restrict__ off)
{
    const int lane  = threadIdx.x & 31;
    const int wid   = threadIdx.x >> 5;
    const int tile  = blockIdx.x * 8 + wid;     // row tile
    const int p_base = tile * 16;

    const int mrow  = lane & 15;
    const int khalf = lane >> 4;

    const int p = p_base + mrow;
    const int n = p >> 16;
    const int s = p & (THW - 1);
    const int t = s >> 12;
    const int h = (s >> 6) & 63;
    const int w = s & 63;

    const float* xTn = xT + ((size_t)n << 20);  // n * THW * C

    int ocl[6];
#pragma unroll
    for (int ct = 0; ct < 6; ++ct) {
        const int o = ct * 16 + mrow;
        ocl[ct] = (o < CO) ? o : (CO - 1);      // clamp; padded cols never stored
    }

    v8f acc[6];
#pragma unroll
    for (int ct = 0; ct < 6; ++ct) acc[ct] = (v8f){};

    for (int kk = 0; kk < RK; kk += 4) {
        const int r  = kk >> 4;                 // tap 0..26, uniform per step
        const int kt = r / 9;
        const int rr = r - kt * 9;
        const int kh = rr / 3;
        const int kw = rr - kh * 3;

        const int tt = t - 1 + kt;
        const int hh = h - 1 + kh;
        const int ww = w - 1 + kw;
        const bool inb = (tt >= 0) && (tt < T) && (hh >= 0) && (hh < H) &&
                         (ww >= 0) && (ww < W);
        const float av = inb ? 1.0f : 0.0f;
        const int tc = tt < 0 ? 0 : (tt > T - 1 ? T - 1 : tt);
        const int hc = hh < 0 ? 0 : (hh > H - 1 ? H - 1 : hh);
        const int wc = ww < 0 ? 0 : (ww > W - 1 ? W - 1 : ww);
        const int sbase = tc * HW + hc * W + wc;

        const int c0 = (kk & 15) + 2 * khalf;   // even -> 8B-aligned pair

        v2f a = *(const v2f*)(xTn + ((size_t)sbase << 4) + c0);
        a.x *= av;
        a.y *= av;

#pragma unroll
        for (int ct = 0; ct < 6; ++ct) {
            const v2f b = *(const v2f*)(w_offT + (ocl[ct] * K3 + r) * 16 + c0);
            acc[ct] = __builtin_amdgcn_wmma_f32_16x16x4_f32(
                false, a, false, b, (short)0, acc[ct], false, false);
        }
    }

#pragma unroll
    for (int ct = 0; ct < 6; ++ct) {
        const int oc = ct * 16 + (lane & 15);
        if (oc < CO) {
            const float bo = b_off[oc];
#pragma unroll
            for (int i = 0; i < 8; ++i) {
                const int pp = p_base + i + 8 * khalf;
                off[pp * CO + oc] = acc[ct][i] + bo;
            }
        }
    }
}

// ---------------------------------------------------------------------------
// Kernel 2: fused trilinear gather + output GEMM (16 positions x 16 out ch).
// Per tap: 8 corner (weight,addr) once per lane; channel-pair b64 gathers.
// ---------------------------------------------------------------------------
__global__ __launch_bounds__(256) void deform_out(const float* __restrict__ xT,
                                                  const float* __restrict__ off,
                                                  const float* __restrict__ wT,
                                                  float* __restrict__ out)
{
    const int lane  = threadIdx.x & 31;
    const int wid   = threadIdx.x >> 5;
    const int tile  = blockIdx.x * 8 + wid;
    const int p_base = tile * 16;

    const int mrow  = lane & 15;
    const int khalf = lane >> 4;

    const int p = p_base + mrow;
    const int n = p >> 16;
    const int s = p & (THW - 1);
    const int t = s >> 12;
    const int h = (s >> 6) & 63;
    const int w = s & 63;

    const float* xTn = xT + ((size_t)n << 20);
    const int ocol  = lane & 15;

    v8f acc = {};

    for (int k = 0; k < K3; ++k) {
        const int kt = k / 9;
        const int rr = k - kt * 9;
        const int kh = rr / 3;
        const int kw = rr - kh * 3;

        const float* op = off + p * CO + k * 3;   // 3 contiguous -> b96
        const float ot = op[0];
        const float oh = op[1];
        const float ow = op[2];

        const float pt = (float)(t - 1 + kt) + ot;
        const float ph = (float)(h - 1 + kh) + oh;
        const float pw = (float)(w - 1 + kw) + ow;

        const float t0 = floorf(pt);
        const float h0 = floorf(ph);
        const float w0 = floorf(pw);

        float wgt[8];
        int   adr[8];                              // byte-ish: element idx << 4
#pragma unroll
        for (int j = 0; j < 8; ++j) {
            const float tcf = t0 + (float)(j >> 2);
            const float hcf = h0 + (float)((j >> 1) & 1);
            const float wcf = w0 + (float)(j & 1);
            float wg = (1.0f - fabsf(pt - tcf)) *
                       (1.0f - fabsf(ph - hcf)) *
                       (1.0f - fabsf(pw - wcf));
            const bool valid = (tcf >= 0.0f) && (tcf < (float)T) &&
                               (hcf >= 0.0f) && (hcf < (float)H) &&
                               (wcf >= 0.0f) && (wcf < (float)W);
            const int ti = (int)fminf(fmaxf(tcf, 0.0f), (float)(T - 1));
            const int hi = (int)fminf(fmaxf(hcf, 0.0f), (float)(H - 1));
            const int wi = (int)fminf(fmaxf(wcf, 0.0f), (float)(W - 1));
            wgt[j] = valid ? wg : 0.0f;            // clip address, mask weight
            adr[j] = (ti * HW + hi * W + wi) << 4; // * C (channel-innermost)
        }

#pragma unroll
        for (int cc = 0; cc < C; cc += 4) {
            const int c0 = cc + 2 * khalf;         // even -> aligned b64 pair
            float a0 = 0.0f, a1 = 0.0f;
#pragma unroll
            for (int j = 0; j < 8; ++j) {
                const v2f g = *(const v2f*)(xTn + adr[j] + c0);
                a0 += wgt[j] * g.x;
                a1 += wgt[j] * g.y;
            }
            v2f a; a.x = a0; a.y = a1;
            const v2f b = *(const v2f*)(wT + (ocol * K3 + k) * 16 + c0);

            acc = __builtin_amdgcn_wmma_f32_16x16x4_f32(
                false, a, false, b, (short)0, acc, false, false);
        }
    }

#pragma unroll
    for (int i = 0; i < 8; ++i) {
        const int pp = p_base + i + 8 * khalf;
        const int nn = pp >> 16;
        const int ss = pp & (THW - 1);
        out[((nn * 16 + ocol) << 16) + ss] = acc[i];
    }
}

extern "C" void kernel_launch(void* const* d_in, const int* in_sizes, int n_in,
                              void* d_out, int out_size, void* d_ws, size_t ws_size,
                              hipStream_t stream) {
    const float* x     = (const float*)d_in[0];   // (2,16,16,64,64)
    const float* w_off = (const float*)d_in[1];   // (81,16,3,3,3)
    const float* b_off = (const float*)d_in[2];   // (81,)
    const float* wmat  = (const float*)d_in[3];   // (16,16,3,3,3)
    float* out = (float*)d_out;                   // (2,16,16,64,64)

    char* ws = (char*)d_ws;
    float* off    = (float*)ws;                   // P*81 floats
    float* xT     = (float*)(ws + XT_OFFSET);     // NTHWC transpose of x
    float* w_offT = (float*)(ws + WOT_OFFSET);    // (81,27,16)
    float* wT     = (float*)(ws + WT_OFFSET);     // (16,27,16)

    (void)in_sizes; (void)n_in; (void)out_size; (void)ws_size;

    // Prep: 2^21 threads cover xT (largest transpose)
    prep<<<(NB * THW * C) / 256, 256, 0, stream>>>(x, w_off, wmat, xT, w_offT, wT);

    // Kernel 1: P/16 = 8192 wave-tiles, 8 waves per 256-thread block
    offset_gemm<<<(P / 16) / 8, 256, 0, stream>>>(xT, w_offT, b_off, off);

    // Kernel 2: same tiling
    deform_out<<<(P / 16) / 8, 256, 0, stream>>>(xT, off, wT, out);
}